// Circuit_28123445854302
// MI455X (gfx1250) — compile-verified
//
#include <hip/hip_runtime.h>

typedef float v2f __attribute__((ext_vector_type(2)));
typedef float v8f __attribute__((ext_vector_type(8)));

#define NT 512          // threads per block (16 wave32)
#define NWAVES 16

// ---------------------------------------------------------------------------
// Structure (compile-time, from _brick_pairs with wire w -> bit 23-w):
//   layer0/2 gates sit on bit pairs (2k+1,2k)  -> aligned with 4-bit blocks [4b+3:4b]
//   layer1/3 gates sit on bit pairs (2k+2,2k+1)-> middle of a P0 block or of an
//                                                 offset block [4k+5:4k+2]
// 22 fused 16x16 block matrices ("slots"):
//   slots  0..5  pass1 (P0, block b): Mid(gate 22-2b) * ( H(gate 10-2b) (x) L(gate 11-2b) )
//   slots  6..10 pass2 (P1, k):       I2 (x) gate(21-2k) (x) I2
//   slots 11..16 pass3 (P0, b):       Mid(gate 45-2b) * ( H(gate 33-2b) (x) L(gate 34-2b) )
//   slots 17..21 pass4 (P1, k):       I2 (x) gate(44-2k) (x) I2
// ---------------------------------------------------------------------------

// gate tensor axes (in0,out0,in1,out1); U4[(a<<1)|b][(i<<1)|j] = g[i*8+a*4+j*2+b]
__device__ __forceinline__ float u4(const float* g, int x, int y) {
  return g[((y >> 1) << 3) + ((x >> 1) << 2) + ((y & 1) << 1) + (x & 1)];
}

__global__ void build_block_mats(const float* __restrict__ gates,
                                 float* __restrict__ mats) {
  int slot = blockIdx.x;
  int tid = threadIdx.x;        // 256 threads: one 16x16 element each
  int r = tid >> 4, c = tid & 15;
  int gL = -1, gH = -1, gM = -1;
  if (slot < 6)       { int b = slot;      gL = 11 - 2*b; gH = 10 - 2*b; gM = 22 - 2*b; }
  else if (slot < 11) { int k = slot - 6;  gM = 21 - 2*k; }
  else if (slot < 17) { int b = slot - 11; gL = 34 - 2*b; gH = 33 - 2*b; gM = 45 - 2*b; }
  else                { int k = slot - 17; gM = 44 - 2*k; }

  float val;
  if (gL >= 0) {
    const float* L  = gates + gL * 16;
    const float* H  = gates + gH * 16;
    const float* Mg = gates + gM * 16;
    // A[r][c] = sum_t Mid16[r][t] * (H (x) L)[t][c];  Mid16 nonzero only when
    // t = (r&8) | (tm<<1) | (r&1)
    float acc = 0.f;
#pragma unroll
    for (int tm = 0; tm < 4; ++tm) {
      int t = (r & 8) | (tm << 1) | (r & 1);
      acc += u4(Mg, (r >> 1) & 3, tm) * u4(H, t >> 2, c >> 2) * u4(L, t & 3, c & 3);
    }
    val = acc;
  } else {
    const float* Mg = gates + gM * 16;
    val = (((r ^ c) & 9) == 0) ? u4(Mg, (r >> 1) & 3, (c >> 1) & 3) : 0.f;
  }
  mats[slot * 256 + r * 16 + c] = val;
}

// ---------------------------------------------------------------------------
// Apply one 16x16 block matrix M along the 4-bit field at LDS-bit position p
// of a 2^16-float LDS-resident chunk, using V_WMMA_F32_16X16X4_F32.
//   out[base + r<<p] = sum_c M[r][c] * s[base + c<<p]
// One wave computes a 16(row) x 16(column) tile per 4 accumulating WMMAs.
// Loops are structurally uniform (compile-time trip counts) so EXEC stays
// all-1s with no per-iteration divergence bookkeeping.
// ---------------------------------------------------------------------------
__device__ __forceinline__ void apply_block(float* s, const float* __restrict__ M,
                                            int p, int lane, int wave) {
  int half = lane >> 4;
  int l16  = lane & 15;
  // A operand (16x4 f32 per K-chunk): lanes 0-15 hold K=0/1 (v0/v1),
  // lanes 16-31 hold K=2/3, M-row = lane%16 (ISA 7.12.2 layout).
  v2f a[4];
#pragma unroll
  for (int kk = 0; kk < 4; ++kk) {
    int c0 = 4 * kk + 2 * half;
    a[kk].x = M[l16 * 16 + c0];
    a[kk].y = M[l16 * 16 + c0 + 1];
  }
  const int lowmask = (1 << p) - 1;
  for (int i = 0; i < 256 / NWAVES; ++i) {       // uniform trip count: 16 tiles/wave
    int t    = wave + i * NWAVES;
    int o    = t * 16 + l16;                     // this lane's column
    int base = ((o & ~lowmask) << 4) | (o & lowmask);
    v8f acc = {0.f, 0.f, 0.f, 0.f, 0.f, 0.f, 0.f, 0.f};
#pragma unroll
    for (int kk = 0; kk < 4; ++kk) {
      int cb = 4 * kk + 2 * half;                // B rows K=cb, cb+1 for this lane half
      v2f b;
      b.x = s[base + (cb << p)];
      b.y = s[base + ((cb + 1) << p)];
      acc = __builtin_amdgcn_wmma_f32_16x16x4_f32(
          /*neg_a=*/false, a[kk], /*neg_b=*/false, b,
          /*c_mod=*/(short)0, acc, /*reuse_a=*/false, /*reuse_b=*/false);
    }
#pragma unroll
    for (int v = 0; v < 8; ++v) {                // D rows v / v+8, col = l16
      int rr = v + 8 * half;
      s[base + (rr << p)] = acc[v];
    }
  }
}

// Stage 1: contiguous chunks over bits [15:0]; synthesize psi0 on the fly,
// apply 12 block-ops in LDS, write full state to d_out.
__global__ void __launch_bounds__(NT)
stage1(const float* __restrict__ wstates,
       const float* __restrict__ mats,
       float* __restrict__ out) {
  extern __shared__ float lds[];
  float* s     = lds;                 // 65536 floats (256 KB)
  float* tabLo = lds + 65536;         // 256
  float* tabHi = lds + 65536 + 256;   // 256
  int tid = threadIdx.x;
  int chunk = blockIdx.x;             // global bits [23:16]

  if (tid < 256) {
    float pl = 1.f, ph = 1.f;
#pragma unroll
    for (int j = 0; j < 8; ++j) {
      pl *= wstates[(23 - j) * 2 + ((tid >> j) & 1)];   // bits 0..7  -> wires 23..16
      ph *= wstates[(15 - j) * 2 + ((tid >> j) & 1)];   // bits 8..15 -> wires 15..8
    }
    tabLo[tid] = pl;
    tabHi[tid] = ph;
  }
  float pref = 1.f;
#pragma unroll
  for (int j = 0; j < 8; ++j)
    pref *= wstates[(7 - j) * 2 + ((chunk >> j) & 1)];  // bits 16..23 -> wires 7..0
  __syncthreads();

#pragma unroll
  for (int i = 0; i < 65536 / NT; ++i) {
    int m = tid + i * NT;
    s[m] = pref * tabHi[(m >> 8) & 255] * tabLo[m & 255];
  }
  __syncthreads();

  int lane = tid & 31, wave = tid >> 5;
  const int slots[12] = {0, 1, 2, 3, 6, 7, 8, 11, 12, 13, 17, 18};
  const int pos[12]   = {0, 4, 8, 12, 2, 6, 10, 0, 4, 8, 2, 6};
  for (int i = 0; i < 12; ++i) {
    apply_block(s, mats + slots[i] * 256, pos[i], lane, wave);
    __syncthreads();
  }

  float4* o4 = (float4*)(out + (size_t)chunk * 65536);
  const float4* s4 = (const float4*)s;
  for (int i = 0; i < 16384 / NT; ++i) {
    int m = tid + i * NT;
    o4[m] = s4[m];
  }
}

// Stage 2: strided chunks over bits [23:10] (x4 consecutive low indices for
// vector loads); apply remaining 10 block-ops in LDS, in place on d_out.
__global__ void __launch_bounds__(NT)
stage2(const float* __restrict__ mats, float* __restrict__ out) {
  extern __shared__ float lds[];
  float* s = lds;                     // 65536 floats
  int tid = threadIdx.x;
  int l0 = blockIdx.x;                // global bits [9:2]
  float4* g4 = (float4*)out;
  // lds[h*4+u] = out[h*1024 + l0*4 + u]  ->  float4 at g4[h*256 + l0]
  for (int i = 0; i < 16384 / NT; ++i) {
    int h = tid + i * NT;
    ((float4*)s)[h] = g4[h * 256 + l0];
  }
  __syncthreads();

  int lane = tid & 31, wave = tid >> 5;
  // LDS-relative positions = (abs bit - 10) + 2 (the x4 tile occupies bits 1:0)
  const int slots[10] = {4, 5, 9, 10, 14, 15, 16, 19, 20, 21};
  const int pos[10]   = {8, 12, 6, 10, 4, 8, 12, 2, 6, 10};
  for (int i = 0; i < 10; ++i) {
    apply_block(s, mats + slots[i] * 256, pos[i], lane, wave);
    __syncthreads();
  }

  for (int i = 0; i < 16384 / NT; ++i) {
    int h = tid + i * NT;
    g4[h * 256 + l0] = ((float4*)s)[h];
  }
}

extern "C" void kernel_launch(void* const* d_in, const int* in_sizes, int n_in,
                              void* d_out, int out_size, void* d_ws, size_t ws_size,
                              hipStream_t stream) {
  (void)in_sizes; (void)n_in; (void)out_size; (void)ws_size;
  const float* states = (const float*)d_in[0];  // (24,2) f32
  const float* gates  = (const float*)d_in[1];  // (46,2,2,2,2) f32
  // d_in[2] (gate_wires) is structural and hardcoded.
  float* out  = (float*)d_out;                  // 2^24 f32
  float* mats = (float*)d_ws;                   // 22 * 256 f32 block matrices

  build_block_mats<<<22, 256, 0, stream>>>(gates, mats);
  stage1<<<256, NT, (65536 + 512) * sizeof(float), stream>>>(states, mats, out);
  stage2<<<256, NT, 65536 * sizeof(float), stream>>>(mats, out);
}